// MultiHeadedAttention_43430709297717
// MI455X (gfx1250) — compile-verified
//
#include <hip/hip_runtime.h>
#include <math.h>

// ---------------------------------------------------------------------------
// MI455X (gfx1250) fused sparse rotary attention.
//   ~1.2 GFLOP fp32, ~6 MB live data -> fully L2-resident (192 MB @ 23.3 TB/s
//   HBM behind it), so the wins are:
//   (a) fp32 WMMA (V_WMMA_F32_16X16X4_F32) for all GEMMs + scores (exact
//       precision match with the fp32 reference),
//   (b) never materializing the 268 MB value5 tensor: rotary is applied only
//       to the k=32 selected keys per (b,h,query) after top-k,
//   (c) CDNA5 async copy (GLOBAL_LOAD_ASYNC_TO_LDS_B128) to stage the W panel
//       shared by all 8 waves of a block into LDS once, instead of 8x
//       redundant global loads of the WMMA A operand.
// ---------------------------------------------------------------------------

typedef __attribute__((ext_vector_type(2))) float v2f;
typedef __attribute__((ext_vector_type(8))) float v8f;

#define B_   4
#define DM   256   // d_model
#define NN   256   // n == m
#define H_   4
#define DIM  64    // d_model / heads
#define KTOP 32
#define LPAD 260   // LDS row pitch (floats): 260 % 64 == 4 -> conflict-free A reads

// D = A(16x4 f32) * B(4x16 f32) + C, wave32.
// A: lane l holds row M=l&15, VGPR0=K(2*(l>>4)), VGPR1=K(2*(l>>4)+1)
// B: lane l holds col N=l&15, VGPR0=row K(2*(l>>4)), VGPR1=K(2*(l>>4)+1)
// C/D: VGPR r holds M = r + 8*(l>>4), N = l&15
__device__ __forceinline__ v8f wmma_f32_4(v2f a, v2f b, v8f c) {
  return __builtin_amdgcn_wmma_f32_16x16x4_f32(
      /*neg_a=*/false, a, /*neg_b=*/false, b,
      /*c_mod=*/(short)0, c, /*reuse_a=*/false, /*reuse_b=*/false);
}

// Async copy of 16B from global to LDS (tracked by ASYNCcnt, no VGPR transit).
__device__ __forceinline__ void async_g2l_b128(unsigned lds_off,
                                               unsigned long long gaddr) {
  asm volatile("global_load_async_to_lds_b128 %0, %1, off"
               :: "v"(lds_off), "v"(gaddr) : "memory");
}
__device__ __forceinline__ void wait_async0() {
  asm volatile("s_wait_asynccnt 0x0" ::: "memory");
}

// ---------------------------------------------------------------------------
// Y[b][o][p] = bias[o] + sum_i W[o][i] * X[b][i][p]     (256x256x256 per batch)
// One wave per 16x16 output tile; the 8 waves of a block share (batch, ot),
// so the 16x256 W panel (16 KB) is async-staged into LDS once per block.
// 1024 tiles -> 128 blocks x 8 waves.
// ---------------------------------------------------------------------------
__global__ __launch_bounds__(256) void proj_gemm_kernel(
    const float* __restrict__ W, const float* __restrict__ bias,
    const float* __restrict__ X, float* __restrict__ Y)
{
  __shared__ float LDSW[16 * LPAD];   // padded 16x256 panel (~16.6 KB)

  const int wid  = threadIdx.x >> 5;
  const int lane = threadIdx.x & 31;
  const int gw   = blockIdx.x * 8 + wid;
  const int batch = gw >> 8;          // uniform across the block
  const int rem   = gw & 255;
  const int ot    = rem >> 4;         // uniform across the block
  const int pt    = rem & 15;
  const int half  = lane >> 4;
  const int l16   = lane & 15;

  // --- stage W panel: rows [ot*16, ot*16+16) x 256 K, 64 B per thread x4 ---
  {
    const int r    = threadIdx.x >> 4;        // panel row 0..15
    const int cseg = (threadIdx.x & 15) * 16; // 16-float segment
    const float* gsrc = W + ((size_t)(ot * 16 + r)) * DM + cseg;
    const unsigned lbase =
        (unsigned)(size_t)&LDSW[r * LPAD + cseg];
    const unsigned long long ga = (unsigned long long)(size_t)gsrc;
#pragma unroll
    for (int jj = 0; jj < 4; ++jj)
      async_g2l_b128(lbase + 16u * jj, ga + 16ull * jj);
    wait_async0();
  }
  __syncthreads();

  const int p = pt * 16 + l16;        // B-col carried by this lane
  const float* __restrict__ Xb = X + (size_t)batch * DM * NN;
  const float* __restrict__ Arow = &LDSW[l16 * LPAD];  // A-row (LDS)

  v8f acc = {};
#pragma unroll 8
  for (int kk = 0; kk < DM; kk += 4) {
    const int k0 = kk + half * 2;
    v2f a;  a.x  = Arow[k0];             a.y  = Arow[k0 + 1];   // ds_load_b64
    v2f bb; bb.x = Xb[k0 * NN + p];      bb.y = Xb[(k0 + 1) * NN + p];
    acc = wmma_f32_4(a, bb, acc);
  }
#pragma unroll
  for (int r = 0; r < 8; ++r) {
    const int om = ot * 16 + r + 8 * half;
    Y[((size_t)batch * DM + om) * NN + pt * 16 + l16] = acc[r] + bias[om];
  }
}

// ---------------------------------------------------------------------------
// Per (b, h, 16-query tile): scores (WMMA) -> LDS, then top-32 + softmax.
// Channel layout of Q/K buffers is [b][d*H+h][point] (matches reshape).
// ---------------------------------------------------------------------------
__global__ __launch_bounds__(256) void scores_topk_kernel(
    const float* __restrict__ Q, const float* __restrict__ Kb,
    float* __restrict__ probs, int* __restrict__ topidx)
{
  __shared__ float S[16 * 256];       // 16 KB score slab
  const int wid  = threadIdx.x >> 5;
  const int lane = threadIdx.x & 31;
  const int blk  = blockIdx.x;
  const int b  = blk >> 6;
  const int h  = (blk >> 4) & 3;
  const int nt = blk & 15;
  const int half = lane >> 4;
  const int l16  = lane & 15;

  const float* __restrict__ Qb  = Q  + (size_t)b * DM * NN;
  const float* __restrict__ Kbb = Kb + (size_t)b * DM * NN;
  const int nrow = nt * 16 + l16;     // query index (A row)

  // --- phase A: 16x256 scores, wave w owns m-tiles 2w, 2w+1 ---
  for (int t = 0; t < 2; ++t) {
    const int mt   = wid * 2 + t;
    const int mcol = mt * 16 + l16;
    v8f acc = {};
#pragma unroll 8
    for (int kk = 0; kk < DIM; kk += 4) {
      const int d0 = kk + half * 2;
      v2f a;  a.x  = Qb[(d0 * H_ + h) * NN + nrow];
              a.y  = Qb[((d0 + 1) * H_ + h) * NN + nrow];
      v2f bb; bb.x = Kbb[(d0 * H_ + h) * NN + mcol];
              bb.y = Kbb[((d0 + 1) * H_ + h) * NN + mcol];
      acc = wmma_f32_4(a, bb, acc);
    }
#pragma unroll
    for (int r = 0; r < 8; ++r)
      S[(r + 8 * half) * 256 + mt * 16 + l16] = acc[r] * 0.125f; // 1/sqrt(64)
  }
  __syncthreads();

  // --- phase B: wave w does rows 2w, 2w+1: iterative top-32 + softmax ---
  for (int rr = 0; rr < 2; ++rr) {
    const int row = wid * 2 + rr;
    float rv[8];
#pragma unroll
    for (int e = 0; e < 8; ++e) rv[e] = S[row * 256 + lane * 8 + e];

    float myv = 0.f; int myi = 0;     // lane t keeps the t-th largest
    for (int it = 0; it < KTOP; ++it) {
      float bv = -INFINITY; int bi = 0;
#pragma unroll
      for (int e = 0; e < 8; ++e)
        if (rv[e] > bv) { bv = rv[e]; bi = lane * 8 + e; }
#pragma unroll
      for (int off = 16; off > 0; off >>= 1) {
        float ov = __shfl_xor(bv, off, 32);
        int   oi = __shfl_xor(bi, off, 32);
        if (ov > bv || (ov == bv && oi < bi)) { bv = ov; bi = oi; }
      }
      if ((bi >> 3) == lane) rv[bi & 7] = -INFINITY;  // mask winner
      if (it == lane)        { myv = bv; myi = bi; }
    }
    const float vmax = __shfl(myv, 0, 32);            // lane0 holds the max
    float e = expf(myv - vmax);
    float s = e;
#pragma unroll
    for (int off = 16; off > 0; off >>= 1) s += __shfl_xor(s, off, 32);
    const float pr = e / s;

    const int    qidx = nt * 16 + row;
    const size_t base = (((size_t)(b * H_ + h) * NN) + qidx) * KTOP;
    probs[base + lane]  = pr;
    topidx[base + lane] = myi;
  }
}

// ---------------------------------------------------------------------------
// Fused rotary + sparse-weighted sum. Wave per (b,h,q); lane t owns the t-th
// top-k key: computes its (alpha,phi) from lrf[key] and rel=x_pos[key]-src_pos[q],
// then a broadcast loop applies rotation to 64 channels (2 per lane).
// out0 written in [b][d*H+h][q] layout == reshape(b, d_model, n).
// ---------------------------------------------------------------------------
__global__ __launch_bounds__(256) void rotary_apply_kernel(
    const float* __restrict__ V, const float* __restrict__ xpos,
    const float* __restrict__ spos, const float* __restrict__ lrf,
    const float* __restrict__ probs, const int* __restrict__ topidx,
    float* __restrict__ out0)
{
  const int wid  = threadIdx.x >> 5;
  const int lane = threadIdx.x & 31;
  const int gw   = blockIdx.x * 8 + wid;
  const int b = gw >> 10;
  const int h = (gw >> 8) & 3;
  const int q = gw & 255;

  const size_t base = (((size_t)(b * H_ + h) * NN) + q) * KTOP;
  const int   j = topidx[base + lane];
  const float p = probs[base + lane];

  // rel = x_pos[j] - src_pos[q], rotated into lrf[j]'s frame: posl_x = L[y][x]*rel_y
  const float rx = xpos[(b * NN + j) * 3 + 0] - spos[(b * NN + q) * 3 + 0];
  const float ry = xpos[(b * NN + j) * 3 + 1] - spos[(b * NN + q) * 3 + 1];
  const float rz = xpos[(b * NN + j) * 3 + 2] - spos[(b * NN + q) * 3 + 2];
  const float* __restrict__ L = lrf + (size_t)(b * NN + j) * 9;
  const float px = L[0] * rx + L[3] * ry + L[6] * rz;
  const float py = L[1] * rx + L[4] * ry + L[7] * rz;
  const float pz = L[2] * rx + L[5] * ry + L[8] * rz;

  float alpha = atanf(py / px);
  if (px < 0.f) alpha += 3.14159265358979323846f;
  float phi = atanf(pz / sqrtf(px * px + py * py));
  if (isnan(alpha)) alpha = 0.f;
  if (isnan(phi))   phi   = 0.f;
  const float cA = cosf(alpha), sA = sinf(alpha);
  const float cP = cosf(phi),   sP = sinf(phi);

  const float* __restrict__ Vb = V + (size_t)b * DM * NN;
  const int d0 = lane, d1 = lane + 32;
  float acc0 = 0.f, acc1 = 0.f;

  for (int t = 0; t < KTOP; ++t) {
    const int   jt  = __shfl(j, t, 32);
    const float pt_ = __shfl(p, t, 32);
    const float cAt = __shfl(cA, t, 32), sAt = __shfl(sA, t, 32);
    const float cPt = __shfl(cP, t, 32), sPt = __shfl(sP, t, 32);
    {
      const float v    = Vb[(d0 * H_ + h) * NN + jt];
      const float vp   = Vb[((d0 ^ 1) * H_ + h) * NN + jt];
      const float desc = (d0 & 1) ? vp : -vp;           // rotate-half pairs
      const float c    = (d0 & 2) ? cPt : cAt;          // d%4<2 -> alpha
      const float s    = (d0 & 2) ? sPt : sAt;
      acc0 = fmaf(pt_, fmaf(v, c, desc * s), acc0);
    }
    {
      const float v    = Vb[(d1 * H_ + h) * NN + jt];
      const float vp   = Vb[((d1 ^ 1) * H_ + h) * NN + jt];
      const float desc = (d1 & 1) ? vp : -vp;
      const float c    = (d1 & 2) ? cPt : cAt;
      const float s    = (d1 & 2) ? sPt : sAt;
      acc1 = fmaf(pt_, fmaf(v, c, desc * s), acc1);
    }
  }
  out0[((size_t)b * DM + (d0 * H_ + h)) * NN + q] = acc0;
  out0[((size_t)b * DM + (d1 * H_ + h)) * NN + q] = acc1;
}

// ---------------------------------------------------------------------------
extern "C" void kernel_launch(void* const* d_in, const int* in_sizes, int n_in,
                              void* d_out, int out_size, void* d_ws, size_t ws_size,
                              hipStream_t stream) {
  (void)in_sizes; (void)n_in; (void)out_size; (void)ws_size;

  const float* x    = (const float*)d_in[0];
  const float* src  = (const float*)d_in[1];
  const float* xpos = (const float*)d_in[2];
  const float* spos = (const float*)d_in[3];
  const float* lrf  = (const float*)d_in[4];
  const float* Wq   = (const float*)d_in[5];
  const float* bq   = (const float*)d_in[6];
  const float* Wk   = (const float*)d_in[7];
  const float* bk   = (const float*)d_in[8];
  const float* Wv   = (const float*)d_in[9];
  const float* bv   = (const float*)d_in[10];
  const float* Wm   = (const float*)d_in[11];
  const float* bm   = (const float*)d_in[12];

  // workspace layout (floats): Q | K | V | probs | topidx | out0  (~5 MB)
  float* ws = (float*)d_ws;
  const size_t proj = (size_t)B_ * DM * NN;        // 262144
  const size_t pk   = (size_t)B_ * H_ * NN * KTOP; // 131072
  float* Qb     = ws;
  float* Kbuf   = Qb + proj;
  float* Vbuf   = Kbuf + proj;
  float* probs  = Vbuf + proj;
  int*   topidx = (int*)(probs + pk);
  float* out0   = (float*)(topidx + pk);

  proj_gemm_kernel<<<128, 256, 0, stream>>>(Wq, bq, x,   Qb);
  proj_gemm_kernel<<<128, 256, 0, stream>>>(Wk, bk, src, Kbuf);
  proj_gemm_kernel<<<128, 256, 0, stream>>>(Wv, bv, src, Vbuf);
  scores_topk_kernel<<<256, 256, 0, stream>>>(Qb, Kbuf, probs, topidx);
  rotary_apply_kernel<<<512, 256, 0, stream>>>(Vbuf, xpos, spos, lrf,
                                               probs, topidx, out0);
  proj_gemm_kernel<<<128, 256, 0, stream>>>(Wm, bm, out0, (float*)d_out);
}